// Mixer_81733227643296
// MI455X (gfx1250) — compile-verified
//
#include <hip/hip_runtime.h>

typedef __attribute__((ext_vector_type(2))) float v2f;
typedef __attribute__((ext_vector_type(4))) float v4f;
typedef __attribute__((ext_vector_type(8))) float v8f;

// D = A(16x4) * B(4x16) + C, all f32, wave32 WMMA.
#define WMMA_F32(Cacc, Afrag, Bfrag)                                          \
  Cacc = __builtin_amdgcn_wmma_f32_16x16x4_f32(false, Afrag, false, Bfrag,    \
                                               (short)0, Cacc, false, false)

// Copy 4KB (one 16-row pairwise tile) global -> LDS with 8 async b128 ops.
// IOFFSET is added to BOTH the per-lane LDS address and the global address,
// so one (lds,vgpr64) pair covers the whole tile. Tracked by ASYNCcnt.
#define ASYNC_TILE_4K(ldsa, gpa)                                                                              \
  do {                                                                                                        \
    asm volatile("global_load_async_to_lds_b128 %0, %1, off" ::"v"(ldsa), "v"(gpa) : "memory");               \
    asm volatile("global_load_async_to_lds_b128 %0, %1, off offset:512" ::"v"(ldsa), "v"(gpa) : "memory");    \
    asm volatile("global_load_async_to_lds_b128 %0, %1, off offset:1024" ::"v"(ldsa), "v"(gpa) : "memory");   \
    asm volatile("global_load_async_to_lds_b128 %0, %1, off offset:1536" ::"v"(ldsa), "v"(gpa) : "memory");   \
    asm volatile("global_load_async_to_lds_b128 %0, %1, off offset:2048" ::"v"(ldsa), "v"(gpa) : "memory");   \
    asm volatile("global_load_async_to_lds_b128 %0, %1, off offset:2560" ::"v"(ldsa), "v"(gpa) : "memory");   \
    asm volatile("global_load_async_to_lds_b128 %0, %1, off offset:3072" ::"v"(ldsa), "v"(gpa) : "memory");   \
    asm volatile("global_load_async_to_lds_b128 %0, %1, off offset:3584" ::"v"(ldsa), "v"(gpa) : "memory");   \
  } while (0)

// Output feature layout per (b, e): [single 0..255 | sum_up 256..511 |
// sum_down 512..767 | pair_up 768..831 | pair_down 832..895]
__global__ __launch_bounds__(128) void mixer_kernel(
    const float* __restrict__ sg_in,   // (B, 16, 256)
    const float* __restrict__ pw_in,   // (B, 256, 64)
    float* __restrict__ out,           // (B, 16, 896)
    int B) {
  // Mask tables stored directly in WMMA A-fragment layout:
  // element [k*64 + lane*2 + v] = MASK[e = lane&15, p = 4k + 2*(lane>>4) + v]
  __shared__ float smu[4096];
  __shared__ float smd[4096];
  // Per-wave double-buffered pairwise tile stage: 4 waves x 2 bufs x 4KB.
  __shared__ __attribute__((aligned(16))) float stage[4][2][1024];

  const int tid = threadIdx.x;
  for (int idx = tid; idx < 4096; idx += 128) {
    const int k = idx >> 6, rem = idx & 63;
    const int L = rem >> 1, v = rem & 1;
    const int e = L & 15;
    const int p = 4 * k + 2 * (L >> 4) + v;
    float up = 0.f, dn = 0.f;
    if (p < 240) {                       // off-diagonal pair block
      const int i = p / 15, r = p - i * 15;
      const int j = (r < i) ? r : r + 1; // column with diagonal removed
      if (i == e) { if (j < 8) up = 1.f; else dn = 1.f; }
    } else {                             // self-entry block
      const int s = p - 240;
      if (s == e) { if (e < 8) up = 1.f; else dn = 1.f; }
    }
    smu[idx] = up;
    smd[idx] = dn;
  }
  __syncthreads();

  const int wave = tid >> 5, lane = tid & 31;
  const int b = blockIdx.x * 4 + wave;   // wave-uniform: no EXEC divergence
  if (b >= B) return;
  const int col = lane & 15, hi = lane >> 4;

  const float* pw = pw_in + (size_t)b * (256 * 64);
  const float* sg = sg_in + (size_t)b * (16 * 256);
  float* ob = out + (size_t)b * (16 * 896);

  // Per-lane base addresses for the async tile copies.
  const unsigned st0 = (unsigned)(uintptr_t)(&stage[wave][0][0]) + (unsigned)lane * 16u;
  const char* pwb = (const char*)pw + (size_t)lane * 16u;

  // Kick off the first two pairwise tiles (groups 0 and 1).
  ASYNC_TILE_4K(st0, pwb);
  ASYNC_TILE_4K(st0 + 4096u, pwb + 4096);

  // ---- single features: streaming copy + spin means (b128 path), overlaps
  // with the in-flight async tile copies ----
  v4f su0 = {}, su1 = {}, sd0 = {}, sd1 = {};
#pragma unroll
  for (int row = 0; row < 16; ++row) {
    const v4f* src = (const v4f*)(sg + row * 256);
    v4f x0 = src[lane];
    v4f x1 = src[lane + 32];
    v4f* dst = (v4f*)(ob + row * 896);
    dst[lane] = x0;
    dst[lane + 32] = x1;
    if (row < 8) { su0 += x0; su1 += x1; } else { sd0 += x0; sd1 += x1; }
  }
  su0 *= 0.125f; su1 *= 0.125f; sd0 *= 0.125f; sd1 *= 0.125f;
#pragma unroll
  for (int row = 0; row < 16; ++row) {
    v4f* du = (v4f*)(ob + row * 896 + 256);
    du[lane] = su0;
    du[lane + 32] = su1;
    v4f* dd = (v4f*)(ob + row * 896 + 512);
    dd[lane] = sd0;
    dd[lane + 32] = sd1;
  }

  // ---- masked pairwise segment-sums: 16x256 mask @ 256x64 via WMMA f32 ----
  // 16 groups of 16 rows; each group = 4 K-chunks of 4 rows. Double-buffered:
  // while computing group g, group g+1 is in flight (ASYNCcnt-tracked).
  v8f cu0 = {}, cu1 = {}, cu2 = {}, cu3 = {};
  v8f cd0 = {}, cd1 = {}, cd2 = {}, cd3 = {};
  for (int g = 0; g < 16; ++g) {
    // Async loads complete in order: <=8 outstanding means group g landed.
    if (g < 15)
      asm volatile("s_wait_asynccnt 0x8" ::: "memory");
    else
      asm volatile("s_wait_asynccnt 0x0" ::: "memory");

    const float* sb = &stage[wave][g & 1][0];
#pragma unroll
    for (int kc = 0; kc < 4; ++kc) {
      const int k = g * 4 + kc;
      v2f au = *(const v2f*)(smu + k * 64 + lane * 2);  // ds_load_b64
      v2f ad = *(const v2f*)(smd + k * 64 + lane * 2);
      // B fragment: N = col, K = 2*hi + v  -> local row 4*kc + 2*hi + v
      const float* p0 = sb + (4 * kc + 2 * hi) * 64 + col;
      v2f b0 = {p0[0], p0[64]};
      v2f b1 = {p0[16], p0[80]};
      v2f b2 = {p0[32], p0[96]};
      v2f b3 = {p0[48], p0[112]};
      WMMA_F32(cu0, au, b0);
      WMMA_F32(cu1, au, b1);
      WMMA_F32(cu2, au, b2);
      WMMA_F32(cu3, au, b3);
      WMMA_F32(cd0, ad, b0);
      WMMA_F32(cd1, ad, b1);
      WMMA_F32(cd2, ad, b2);
      WMMA_F32(cd3, ad, b3);
    }
    // Refill the buffer we just finished with group g+2. Hardware-order the
    // refill after ALL of this group's LDS reads have completed (WAR guard):
    // the async engine must not overwrite words an in-flight ds_load still
    // needs. The WMMAs above already forced most of these waits, so this is
    // nearly free.
    if (g + 2 < 16) {
      asm volatile("s_wait_dscnt 0x0" ::: "memory");
      const unsigned ldsa = st0 + (unsigned)((g & 1) * 4096);
      const char* gpa = pwb + (size_t)(g + 2) * 4096;
      ASYNC_TILE_4K(ldsa, gpa);
    }
  }

  // C/D layout: VGPR r -> electron r (lanes 0-15) or r+8 (lanes 16-31), N = col
#pragma unroll
  for (int r = 0; r < 8; ++r) {
    const int e = hi ? (r + 8) : r;
    float* po = ob + e * 896 + col;
    po[768 + 0]  = cu0[r] * 0.125f;
    po[768 + 16] = cu1[r] * 0.125f;
    po[768 + 32] = cu2[r] * 0.125f;
    po[768 + 48] = cu3[r] * 0.125f;
    po[832 + 0]  = cd0[r] * 0.125f;
    po[832 + 16] = cd1[r] * 0.125f;
    po[832 + 32] = cd2[r] * 0.125f;
    po[832 + 48] = cd3[r] * 0.125f;
  }
}

extern "C" void kernel_launch(void* const* d_in, const int* in_sizes, int n_in,
                              void* d_out, int out_size, void* d_ws, size_t ws_size,
                              hipStream_t stream) {
  const float* single = (const float*)d_in[0];    // (B, 16, 256) f32
  const float* pairwise = (const float*)d_in[1];  // (B, 256, 64) f32
  float* out = (float*)d_out;                     // (B, 16, 896) f32
  const int B = in_sizes[0] / (16 * 256);
  const int blocks = (B + 3) / 4;                 // 4 waves per block, 1 batch per wave
  mixer_kernel<<<blocks, 128, 0, stream>>>(single, pairwise, out, B);
}